// DetectHead_33036888441039
// MI455X (gfx1250) — compile-verified
//
#include <hip/hip_runtime.h>
#include <hip/hip_bf16.h>
#include <stdint.h>

// Problem constants (fixed by the reference)
#define NB      16
#define NCH     22          // 6 + CLS_NUM
#define CLS     16
#define NH      160
#define NW      160
#define HW      (NH * NW)   // 25600
#define NTGT    2048
#define TILE    256         // pixels per block
#define NTILES  (HW / TILE) // 100
#define EPSL    1e-7f

typedef uint32_t v4u __attribute__((ext_vector_type(4)));
typedef int32_t  v8i __attribute__((ext_vector_type(8)));
typedef int32_t  v4i __attribute__((ext_vector_type(4)));

__device__ __forceinline__ float sigm(float z) {
    return 1.0f / (1.0f + __expf(-z));
}

// 5-step wave32 butterfly sum
__device__ __forceinline__ float wave_sum(float v) {
#pragma unroll
    for (int s = 16; s > 0; s >>= 1) v += __shfl_xor(v, s, 32);
    return v;
}

// ---------------------------------------------------------------------------
// Kernel 0: zero the mask bit-arrays and the per-class accumulators.
// ---------------------------------------------------------------------------
__global__ void dh_init(uint32_t* __restrict__ ws, int n) {
    int i = blockIdx.x * blockDim.x + threadIdx.x;
    if (i < n) ws[i] = 0u;
}

// ---------------------------------------------------------------------------
// Kernel 1: rasterize rotated boxes into obj / wide bitmasks.
// grid = (NTILES, NTGT), block = 256. Each wave covers 32 consecutive grid
// cells == exactly one bitmask word -> ballot + single atomicOr per wave.
// ---------------------------------------------------------------------------
__global__ void dh_raster(const float* __restrict__ tgt,
                          uint32_t* __restrict__ obj_bits,
                          uint32_t* __restrict__ wide_bits) {
    const int n   = blockIdx.y;
    const int tid = threadIdx.x;

    __shared__ float tp[7];
    if (tid < 7) tp[tid] = tgt[n * 7 + tid];
    __syncthreads();

    const int   bI  = (int)tp[0];
    const int   cI  = (int)tp[1];
    const float cx  = tp[2] * 0.125f;   // / SCALE
    const float cy  = tp[3] * 0.125f;
    const float w   = tp[4] * 0.125f;
    const float h   = tp[5] * 0.125f;
    float sn, cs;
    __sincosf(tp[6], &sn, &cs);

    const float xr = cx * cs + cy * sn;
    const float yr = -cx * sn + cy * cs;
    const float x3 = xr - h * 0.5f, x4 = xr + h * 0.5f;
    const float y3 = yr - w * 0.5f, y4 = yr + w * 0.5f;
    const float x1 = x3 - 0.5f, x2 = x4 + 0.5f;
    const float y1 = y3 - 0.5f, y2 = y4 + 0.5f;

    const int   p  = blockIdx.x * TILE + tid;
    const float gx = (float)(p % NW) + 0.5f;
    const float gy = (float)(p / NW) + 0.5f;
    const float vx = cs * gx + sn * gy;
    const float vy = -sn * gx + cs * gy;

    const bool obj  = (x3 <= vx) && (vx <= x4) && (y3 <= vy) && (vy <= y4);
    const bool wide = (x1 <= vx) && (vx <= x2) && (y1 <= vy) && (vy <= y2);

    const uint32_t mo = __builtin_amdgcn_ballot_w32(obj);
    const uint32_t mw = __builtin_amdgcn_ballot_w32(wide);

    if ((tid & 31) == 0) {
        const uint32_t word =
            (((uint32_t)(cI * NB + bI)) * (uint32_t)HW + (uint32_t)p) >> 5;
        if (mo) atomicOr(&obj_bits[word], mo);
        if (mw) atomicOr(&wide_bits[word], mw);
    }
}

// ---------------------------------------------------------------------------
// Kernel 2: fused loss accumulation + box decode.
// grid = NB * NTILES blocks, 256 threads. The 22 strided channel rows for the
// block's 256-pixel tile are DMA'd into LDS with one TDM descriptor.
// Loss partials are wave-reduced (ballot/popc + shfl_xor tree), so LDS
// atomics drop to one per wave per class.
// ---------------------------------------------------------------------------
__global__ void dh_loss_decode(const float* __restrict__ predict,
                               const uint32_t* __restrict__ obj_bits,
                               const uint32_t* __restrict__ wide_bits,
                               float* __restrict__ gacc,
                               float* __restrict__ dout) {
    const int tid   = threadIdx.x;
    const int lane  = tid & 31;
    const int b     = blockIdx.x / NTILES;
    const int tile0 = blockIdx.x % NTILES;
    const int pix0  = tile0 * TILE;
    const int p     = pix0 + tid;

    __shared__ float sh[NCH * TILE];   // [channel][pixel] tile, filled by TDM
    __shared__ float acc[64];          // loss_obj[16] loss_noobj[16] n_obj[16] n_noobj[16]
    __shared__ float outb[TILE * 7];   // coalescing stage for the boxes

    if (tid < 64) acc[tid] = 0.0f;

    // --- Tensor Data Mover: 2-D tile (256 x 22ch, 4B elems, row stride 25600)
    if (tid < 32) {                    // one wave issues the DMA (EXEC ignored)
        const float* gsrc =
            predict + (size_t)b * (size_t)(NCH * HW) + (size_t)pix0;
        const uint64_t ga  = (uint64_t)(uintptr_t)gsrc;
        const uint32_t lds = (uint32_t)(uintptr_t)(&sh[0]); // LDS byte offset

        v4u g0 = { 1u,                                  // count=1, user mode
                   lds,                                 // lds_addr
                   (uint32_t)ga,                        // global_addr[31:0]
                   (uint32_t)((ga >> 32) & 0x1FFFFFFu)  // global_addr[56:32]
                       | (2u << 30) };                  // type=2 (image)
        v8i g1 = { (int)(2u << 16),           // data_size=2 -> 4 bytes
                   (int)(0x6400u << 16),      // tensor_dim0[15:0]=25600
                   (int)((uint32_t)NCH << 16),// tensor_dim1[15:0]=22
                   (int)(256u << 16),         // tile_dim0=256
                   NCH,                       // tile_dim1=22, tile_dim2=0
                   HW,                        // tensor_dim0_stride low32 = 25600
                   0, 0 };
        v4i g2 = { 0, 0, 0, 0 };
        v4i g3 = { 0, 0, 0, 0 };
        v8i g4 = { 0, 0, 0, 0, 0, 0, 0, 0 };
        __builtin_amdgcn_tensor_load_to_lds(g0, g1, g2, g3, g4, 0);
        __builtin_amdgcn_s_wait_tensorcnt(0);
    }
    __syncthreads();

    // --- decode inputs
    const float x  = sigm(sh[0 * TILE + tid]) * 5.0f - 2.0f;
    const float y  = sigm(sh[1 * TILE + tid]) * 5.0f - 2.0f;
    const float zw = sh[2 * TILE + tid];
    const float zh = sh[3 * TILE + tid];
    const float pp = sigm(sh[4 * TILE + tid]) * 2.0f - 1.0f;
    const float qq = sigm(sh[5 * TILE + tid]) * 2.0f - 1.0f;

    // --- per-class: conf max/argmax + masked BCE, wave-level reduction
    float best  = -1.0f;
    float bestc = 0.0f;
    const uint32_t bit = 1u << (p & 31);
#pragma unroll
    for (int c = 0; c < CLS; ++c) {
        const float sc = sigm(sh[(6 + c) * TILE + tid]);
        if (sc > best) { best = sc; bestc = (float)c; }

        const float confc = fminf(fmaxf(sc, EPSL), 1.0f - EPSL);
        const uint32_t word =
            (((uint32_t)(c * NB + b)) * (uint32_t)HW + (uint32_t)p) >> 5;
        const bool obj = (obj_bits[word] & bit) != 0;
        const bool noo = (wide_bits[word] & bit) == 0;

        const uint32_t mo = __builtin_amdgcn_ballot_w32(obj);
        const uint32_t mn = __builtin_amdgcn_ballot_w32(noo);

        if (mo) {                               // wave-uniform; usually false
            const float v = wave_sum(obj ? -__logf(confc) : 0.0f);
            if (lane == 0) {
                atomicAdd(&acc[c], v);
                atomicAdd(&acc[32 + c], (float)__popc(mo));
            }
        }
        if (mn) {                               // wave-uniform; usually true
            const float v = wave_sum(noo ? -__logf(1.0f - confc) : 0.0f);
            if (lane == 0) {
                atomicAdd(&acc[16 + c], v);
                atomicAdd(&acc[48 + c], (float)__popc(mn));
            }
        }
    }

    // --- box decode
    const float t    = sqrtf((1.0f + pp) * 0.5f + 1e-16f);
    const float bang = atan2f(qq / (2.0f * t + 1e-16f), t);
    const float gxf  = (float)(p % NW);
    const float gyf  = (float)(p / NW);

    outb[tid * 7 + 0] = best;
    outb[tid * 7 + 1] = bestc;
    outb[tid * 7 + 2] = (gxf + x) * 8.0f;
    outb[tid * 7 + 3] = (gyf + y) * 8.0f;
    outb[tid * 7 + 4] = __expf(zw) * 8.0f;
    outb[tid * 7 + 5] = __expf(zh) * 8.0f;
    outb[tid * 7 + 6] = bang;
    __syncthreads();

    // coalesced store of the 256*7 tile (output tuple: [loss, boxes...])
    float* dst = dout + 1 + ((size_t)b * HW + (size_t)pix0) * 7;
    for (int i = tid; i < TILE * 7; i += TILE) dst[i] = outb[i];

    if (tid < 64) atomicAdd(&gacc[tid], acc[tid]);
}

// ---------------------------------------------------------------------------
// Kernel 3: fold per-class sums into the scalar loss.
// ---------------------------------------------------------------------------
__global__ void dh_finalize(const float* __restrict__ gacc,
                            float* __restrict__ dout) {
    if (threadIdx.x == 0 && blockIdx.x == 0) {
        float s = 0.0f;
        for (int c = 0; c < CLS; ++c) {
            const float lo = gacc[c];
            const float ln = gacc[16 + c];
            const float no = gacc[32 + c];
            const float nn = gacc[48 + c];
            if (no > 0.0f)
                s += lo / fmaxf(no, 1.0f) + ln / fmaxf(nn, 1.0f);
        }
        dout[0] = s;
    }
}

// ---------------------------------------------------------------------------
extern "C" void kernel_launch(void* const* d_in, const int* in_sizes, int n_in,
                              void* d_out, int out_size, void* d_ws, size_t ws_size,
                              hipStream_t stream) {
    (void)in_sizes; (void)n_in; (void)out_size; (void)ws_size;

    const float* predict = (const float*)d_in[0]; // (16, 22, 160, 160)
    const float* targets = (const float*)d_in[1]; // (2048, 7)
    float* out = (float*)d_out;                   // [loss, boxes(16*25600*7)]

    const int mask_words = CLS * NB * (HW / 32);  // 204800 words per mask
    uint32_t* obj_bits  = (uint32_t*)d_ws;
    uint32_t* wide_bits = obj_bits + mask_words;
    float*    gacc      = (float*)(wide_bits + mask_words); // 64 floats

    const int zn = mask_words * 2 + 64;
    dh_init<<<dim3((zn + 255) / 256), dim3(256), 0, stream>>>((uint32_t*)d_ws, zn);

    dh_raster<<<dim3(NTILES, NTGT), dim3(256), 0, stream>>>(
        targets, obj_bits, wide_bits);

    dh_loss_decode<<<dim3(NB * NTILES), dim3(256), 0, stream>>>(
        predict, obj_bits, wide_bits, gacc, out);

    dh_finalize<<<dim3(1), dim3(64), 0, stream>>>(gacc, out);
}